// Encoder_35270271435370
// MI455X (gfx1250) — compile-verified
//
#include <hip/hip_runtime.h>
#include <hip/hip_bf16.h>

#define E_SIZE 6884
#define R_SIZE 990
#define DIM 128
#define HID 64
#define NUM_EDGES 250000

typedef __attribute__((ext_vector_type(2))) float v2f;
typedef __attribute__((ext_vector_type(8))) float v8f;

// ---------------------------------------------------------------------------
// Count edges per (dst, relation) bucket:  cnt[dst*R + type] += 1
// ---------------------------------------------------------------------------
__global__ void count_kernel(const int* __restrict__ dst,
                             const int* __restrict__ etype,
                             unsigned int* __restrict__ cnt, int E) {
    int e = blockIdx.x * blockDim.x + threadIdx.x;
    if (e < E) {
        atomicAdd(&cnt[dst[e] * R_SIZE + etype[e]], 1u);
    }
}

// ---------------------------------------------------------------------------
// Generic fp32 WMMA GEMM: C[M x 16tiles] = A[M x K] @ B[K x ncols] (+bias)
// One wave (32 lanes) computes one 16x16 output tile using
// V_WMMA_F32_16X16X4_F32 with a K-loop of step 4.
// A fragment (16x4, ISA 7.12.2): lane m=lane&15, half=lane>>4;
//   vgpr0 = A[m, k+2*half], vgpr1 = A[m, k+2*half+1]
// B fragment (4x16): mirrored K-major layout.
// C/D: vgpr v -> row v + 8*half, col = lane&15.
// ---------------------------------------------------------------------------
__global__ void wmma_gemm_f32(const float* __restrict__ A, int lda,
                              const float* __restrict__ B, int ldb,
                              float* __restrict__ C, int ldc,
                              const float* __restrict__ bias,
                              int M, int K) {
    const int row0 = blockIdx.x * 16;
    const int col0 = blockIdx.y * 16;
    const int lane = threadIdx.x;          // 0..31
    const int m    = lane & 15;
    const int half = lane >> 4;

    int arow = row0 + m;
    if (arow >= M) arow = M - 1;           // clamp (keeps EXEC all-ones; row m of
                                           // D depends only on row m of A)
    const int bcol = col0 + m;

    v8f acc = {};
    for (int k = 0; k < K; k += 4) {
        v2f a, b;
        a.x = A[arow * lda + k + 2 * half];
        a.y = A[arow * lda + k + 2 * half + 1];
        b.x = B[(k + 2 * half) * ldb + bcol];
        b.y = B[(k + 2 * half + 1) * ldb + bcol];
        acc = __builtin_amdgcn_wmma_f32_16x16x4_f32(
            /*neg_a=*/false, a, /*neg_b=*/false, b,
            /*c_mod=*/(short)0, acc, /*reuse_a=*/false, /*reuse_b=*/false);
    }

    const float bv = bias ? bias[col0 + m] : 0.0f;
#pragma unroll
    for (int v = 0; v < 8; ++v) {
        int r = row0 + v + 8 * half;
        if (r < M) {
            C[r * ldc + col0 + m] = acc[v] + bv;
        }
    }
}

// ---------------------------------------------------------------------------
// conv1 edge messages: m1 = comp1[t,0]*xb[src,0,:] + comp1[t,1]*xb[src,1,:]
// agg1[dst,:] += m1 * norm          (64 threads per edge)
// ---------------------------------------------------------------------------
__global__ void edge1_kernel(const int* __restrict__ src,
                             const int* __restrict__ dst,
                             const int* __restrict__ etype,
                             const unsigned int* __restrict__ cnt,
                             const float* __restrict__ comp1,
                             const float* __restrict__ xb,
                             float* __restrict__ agg1, int E) {
    int tid = blockIdx.x * 256 + threadIdx.x;
    int e = tid >> 6;
    int o = tid & 63;
    if (e >= E) return;
    const int s = src[e], d = dst[e], t = etype[e];
    unsigned int c = cnt[d * R_SIZE + t];
    float norm = 1.0f / (float)(c ? c : 1u);
    float c0 = comp1[t * 2 + 0];
    float c1 = comp1[t * 2 + 1];
    float m1 = c0 * xb[s * 128 + o] + c1 * xb[s * 128 + 64 + o];
    atomicAdd(&agg1[d * HID + o], m1 * norm);
}

// ---------------------------------------------------------------------------
// conv2 edge messages (block-diagonal, per-edge weights -> VALU FMA):
// 128 threads per edge; thread owns (block b, out col dcol): 16-MAC dot.
// W2 slice is L2-resident (8.1 MB total).
// ---------------------------------------------------------------------------
__global__ void edge2_kernel(const int* __restrict__ src,
                             const int* __restrict__ dst,
                             const int* __restrict__ etype,
                             const unsigned int* __restrict__ cnt,
                             const float* __restrict__ h,
                             const float* __restrict__ W2,
                             float* __restrict__ agg2, int E) {
    int tid  = blockIdx.x * 256 + threadIdx.x;
    int e    = tid >> 7;
    int r    = tid & 127;
    int b    = r >> 5;
    int dcol = r & 31;
    if (e >= E) return;
    const int s = src[e], d = dst[e], t = etype[e];
    unsigned int c = cnt[d * R_SIZE + t];
    float norm = 1.0f / (float)(c ? c : 1u);
    const float* hb = h + s * HID + b * 16;
    const float* w  = W2 + (t * 4 + b) * (16 * 32) + dcol;
    float acc = 0.0f;
#pragma unroll
    for (int cc = 0; cc < 16; ++cc)
        acc = fmaf(hb[cc], w[cc * 32], acc);
    atomicAdd(&agg2[d * DIM + b * 32 + dcol], acc * norm);
}

// ---------------------------------------------------------------------------
__global__ void relu_kernel(const float* __restrict__ in,
                            float* __restrict__ out, int n) {
    int i = blockIdx.x * 256 + threadIdx.x;
    if (i < n) out[i] = fmaxf(in[i], 0.0f);
}

// ---------------------------------------------------------------------------
extern "C" void kernel_launch(void* const* d_in, const int* in_sizes, int n_in,
                              void* d_out, int out_size, void* d_ws, size_t ws_size,
                              hipStream_t stream) {
    const float* emb    = (const float*)d_in[0];   // [N,128]
    const float* basis1 = (const float*)d_in[1];   // [2,128,64]
    const float* comp1  = (const float*)d_in[2];   // [990,2]
    const float* root1  = (const float*)d_in[3];   // [128,64]
    const float* bias1  = (const float*)d_in[4];   // [64]
    const float* W2     = (const float*)d_in[5];   // [990,4,16,32]
    const float* root2  = (const float*)d_in[6];   // [64,128]
    const float* bias2  = (const float*)d_in[7];   // [128]
    const int*   eidx   = (const int*)d_in[8];     // [2,E]
    const int*   etype  = (const int*)d_in[9];     // [E]
    const int*   src = eidx;
    const int*   dst = eidx + NUM_EDGES;
    float* out = (float*)d_out;

    // Workspace layout (all 4-byte elements), ~38 MB total:
    unsigned int* cnt = (unsigned int*)d_ws;                  // N*R u32
    size_t off = (size_t)E_SIZE * R_SIZE;
    float* base = (float*)d_ws;
    float* xb   = base + off;  off += (size_t)E_SIZE * 128;   // [N,128] (b*64+o)
    float* agg1 = base + off;  off += (size_t)E_SIZE * HID;   // [N,64]
    float* h    = base + off;  off += (size_t)E_SIZE * HID;   // [N,64]
    float* agg2 = base + off;                                 // [N,128]

    // 1) per-(dst, relation) counts
    hipMemsetAsync(cnt, 0, sizeof(unsigned int) * (size_t)E_SIZE * R_SIZE, stream);
    count_kernel<<<(NUM_EDGES + 255) / 256, 256, 0, stream>>>(dst, etype, cnt, NUM_EDGES);

    // 2) conv1 dense parts via fp32 WMMA:
    //    xb[:,0:64] = emb@basis1[0]; xb[:,64:128] = emb@basis1[1];
    //    agg1 seeded with emb@root1 + bias1
    const int mt = (E_SIZE + 15) / 16;
    dim3 g1(mt, HID / 16);
    wmma_gemm_f32<<<g1, 32, 0, stream>>>(emb, DIM, basis1,             HID, xb,       128, nullptr, E_SIZE, DIM);
    wmma_gemm_f32<<<g1, 32, 0, stream>>>(emb, DIM, basis1 + DIM * HID, HID, xb + HID, 128, nullptr, E_SIZE, DIM);
    wmma_gemm_f32<<<g1, 32, 0, stream>>>(emb, DIM, root1,              HID, agg1,     HID, bias1,   E_SIZE, DIM);

    // 3) conv1 edge scatter (mean-normalized), then h = relu(agg1)
    edge1_kernel<<<(NUM_EDGES * 64 + 255) / 256, 256, 0, stream>>>(
        src, dst, etype, cnt, comp1, xb, agg1, NUM_EDGES);
    relu_kernel<<<(E_SIZE * HID + 255) / 256, 256, 0, stream>>>(agg1, h, E_SIZE * HID);

    // 4) conv2 dense part via WMMA: agg2 seeded with h@root2 + bias2
    dim3 g2(mt, DIM / 16);
    wmma_gemm_f32<<<g2, 32, 0, stream>>>(h, HID, root2, DIM, agg2, DIM, bias2, E_SIZE, HID);

    // 5) conv2 edge scatter, then out = relu(agg2)
    edge2_kernel<<<(NUM_EDGES * 128 + 255) / 256, 256, 0, stream>>>(
        src, dst, etype, cnt, h, W2, agg2, NUM_EDGES);
    relu_kernel<<<(E_SIZE * DIM + 255) / 256, 256, 0, stream>>>(agg2, out, E_SIZE * DIM);
}